// AxialAttention_3D_16183436771877
// MI455X (gfx1250) — compile-verified
//
#include <hip/hip_runtime.h>

// ---------------------------------------------------------------------------
// AxialAttention-3D forward for MI455X (gfx1250), full fp32 via
// V_WMMA_F32_16X16X4_F32 (exact fp32 semantics; problem is HBM-bound so
// low-precision WMMA buys nothing here).
//
// Pipeline (all on `stream`, deterministic, no atomics):
//   K1: qkv(128,P) = W(128,64) @ X(64,P)                     [WMMA, K=64]
//   K2: per-channel mean/var of qkv -> scale/shift
//   K3: per-(b,head): qk (WMMA, K=4) + qr + kr -> per-block stat partials
//   K4: reduce partials -> BN-sim scale/shift
//   K5: full attention: qk(WMMA)+qr+kr -> BN -> softmax -> sv(WMMA,K=56)/sve
//   K6: per-channel stats of so (reuses K2 kernel)
//   K7: BN-out, pairwise channel sum, write (c,d,h,w) output
// ---------------------------------------------------------------------------

#define KW     56
#define CIN    64
#define OC2    128
#define NHEAD  8
#define BATCH  1792                      // 32*56
#define PTOT   (BATCH * KW)              // 100352
#define RELW   111
#define F_QR   0.1f
#define F_KR   0.1f
#define F_SVE  0.1f
#define F_SV   1.0f
#define EPSB   1e-5f

typedef __attribute__((ext_vector_type(2))) float v2f;
typedef __attribute__((ext_vector_type(8))) float v8f;

__device__ __forceinline__ v8f wmma_f32_k4(v2f a, v2f b, v8f c) {
  // D(16x16) += A(16x4) * B(4x16), full fp32
  return __builtin_amdgcn_wmma_f32_16x16x4_f32(false, a, false, b, (short)0, c,
                                               false, false);
}

// ---------------------------------------------------------------------------
// K1: qkv = W @ X.  grid = (P/64, 128/16), block = 128 (4 waves).
// Wave w computes the 16x16 tile at (m0, blockIdx.x*64 + w*16); 16 k-steps.
// A-frag layout: lane(0-15) m=lane, K={0,1}; lane(16-31) K={2,3} (ISA 7.12.2)
// ---------------------------------------------------------------------------
__global__ __launch_bounds__(128)
void k1_qkv_gemm(const float* __restrict__ x, const float* __restrict__ w,
                 float* __restrict__ qkv) {
  const int lane = threadIdx.x & 31;
  const int wave = threadIdx.x >> 5;
  const int m0 = blockIdx.y * 16;
  const int n0 = blockIdx.x * 64 + wave * 16;
  const int mn = lane & 15;      // A row / B col / D col held by this lane
  const int kh = lane >> 4;      // which K-half of the 4-wide contraction
  v8f acc = {};
  #pragma unroll
  for (int ks = 0; ks < 16; ++ks) {
    const int kk = ks * 4 + 2 * kh;
    v2f a, b;
    a.x = w[(m0 + mn) * CIN + kk];
    a.y = w[(m0 + mn) * CIN + kk + 1];
    b.x = x[(size_t)kk * PTOT + n0 + mn];
    b.y = x[(size_t)(kk + 1) * PTOT + n0 + mn];
    acc = wmma_f32_k4(a, b, acc);
  }
  #pragma unroll
  for (int v = 0; v < 8; ++v) {
    const int m = v + 8 * kh;    // D row (ISA C/D layout)
    qkv[(size_t)(m0 + m) * PTOT + n0 + mn] = acc[v];
  }
}

// ---------------------------------------------------------------------------
// K2/K6: per-channel sum/sumsq over P values -> BN scale/shift.
// One block per channel; fixed-order tree reduction (deterministic).
// ---------------------------------------------------------------------------
__global__ __launch_bounds__(256)
void k2_chstats(const float* __restrict__ src, const float* __restrict__ gma,
                const float* __restrict__ bta, float* __restrict__ bn) {
  const int ch = blockIdx.x;
  __shared__ float ssum[256], ssq[256];
  float s = 0.f, q = 0.f;
  for (int p = threadIdx.x; p < PTOT; p += 256) {
    const float v = src[(size_t)ch * PTOT + p];
    s += v; q += v * v;
  }
  ssum[threadIdx.x] = s; ssq[threadIdx.x] = q;
  __syncthreads();
  for (int off = 128; off > 0; off >>= 1) {
    if (threadIdx.x < off) {
      ssum[threadIdx.x] += ssum[threadIdx.x + off];
      ssq[threadIdx.x]  += ssq[threadIdx.x + off];
    }
    __syncthreads();
  }
  if (threadIdx.x == 0) {
    const float mean = ssum[0] / (float)PTOT;
    const float var  = ssq[0] / (float)PTOT - mean * mean;
    const float sc   = gma[ch] * rsqrtf(var + EPSB);
    bn[2 * ch]     = sc;
    bn[2 * ch + 1] = bta[ch] - mean * sc;
  }
}

// ---------------------------------------------------------------------------
// K3: per-(b,g) recompute qk (WMMA, K=4) / qr / kr, emit per-block
// (sum,sumsq) partials for the 3 BN-sim channels of this head.
// Zero padding of q/k columns 56..63 makes padded qk entries exactly 0,
// so they contribute nothing to sum or sumsq (count handled analytically).
// ---------------------------------------------------------------------------
__global__ __launch_bounds__(128)
void k3_sim_stats(const float* __restrict__ qkv, const float* __restrict__ bnq,
                  const float* __restrict__ relative, float* __restrict__ part) {
  __shared__ float qs[4 * 64];
  __shared__ float ks[4 * 64];
  __shared__ float rl[8 * RELW];
  __shared__ float red[128];
  const int tid = threadIdx.x;
  const int bg = blockIdx.x;
  const int b = bg >> 3, g = bg & 7;

  for (int idx = tid; idx < 4 * 64; idx += 128) {
    const int c = idx >> 6, i = idx & 63;
    const int oq = g * 16 + c, ok = g * 16 + 4 + c;
    float vq = 0.f, vk = 0.f;
    if (i < KW) {
      vq = bnq[2 * oq] * qkv[(size_t)oq * PTOT + b * KW + i] + bnq[2 * oq + 1];
      vk = bnq[2 * ok] * qkv[(size_t)ok * PTOT + b * KW + i] + bnq[2 * ok + 1];
    }
    qs[idx] = vq; ks[idx] = vk;
  }
  for (int idx = tid; idx < 8 * RELW; idx += 128) rl[idx] = relative[idx];
  __syncthreads();

  const int lane = tid & 31, wave = tid >> 5;
  const int mn = lane & 15, kh = lane >> 4;
  const int j0 = wave * 16;
  float qk_s = 0.f, qk_q = 0.f;
  for (int ti = 0; ti < 4; ++ti) {
    const int i0 = ti * 16;
    v2f a, bb;
    a.x  = qs[(2 * kh + 0) * 64 + i0 + mn];
    a.y  = qs[(2 * kh + 1) * 64 + i0 + mn];
    bb.x = ks[(2 * kh + 0) * 64 + j0 + mn];
    bb.y = ks[(2 * kh + 1) * 64 + j0 + mn];
    v8f d = {};
    d = wmma_f32_k4(a, bb, d);
    #pragma unroll
    for (int v = 0; v < 8; ++v) { const float t = d[v]; qk_s += t; qk_q += t * t; }
  }

  float qr_s = 0.f, qr_q = 0.f, kr_s = 0.f, kr_q = 0.f;
  for (int e = tid; e < KW * KW; e += 128) {
    const int i = e / KW, j = e - i * KW;
    const int dij = i - j + (KW - 1);
    const int dji = 110 - dij;
    float qr = qs[0 * 64 + i] * rl[0 * RELW + dij]
             + qs[1 * 64 + i] * rl[1 * RELW + dij]
             + qs[2 * 64 + i] * rl[2 * RELW + dij]
             + qs[3 * 64 + i] * rl[3 * RELW + dij];
    qr *= F_QR;
    float kr = ks[0 * 64 + j] * rl[4 * RELW + dji]
             + ks[1 * 64 + j] * rl[5 * RELW + dji]
             + ks[2 * 64 + j] * rl[6 * RELW + dji]
             + ks[3 * 64 + j] * rl[7 * RELW + dji];
    kr *= F_KR;
    qr_s += qr; qr_q += qr * qr; kr_s += kr; kr_q += kr * kr;
  }

  float vals[6] = {qk_s, qk_q, qr_s, qr_q, kr_s, kr_q};
  for (int t = 0; t < 6; ++t) {
    red[tid] = vals[t];
    __syncthreads();
    for (int off = 64; off > 0; off >>= 1) {
      if (tid < off) red[tid] += red[tid + off];
      __syncthreads();
    }
    if (tid == 0) part[(size_t)bg * 6 + t] = red[0];
    __syncthreads();
  }
}

// ---------------------------------------------------------------------------
// K4: reduce per-block partials -> BN-sim scale/shift (24 channels)
// ---------------------------------------------------------------------------
__global__ void k4_bns(const float* __restrict__ part, const float* __restrict__ gma,
                       const float* __restrict__ bta, float* __restrict__ bns) {
  const int ch = threadIdx.x;
  if (ch >= 24) return;
  const int t = ch >> 3, g = ch & 7;  // t: 0=qk 1=qr 2=kr
  float s = 0.f, q = 0.f;
  for (int blk = 0; blk < BATCH; ++blk) {
    const float* p = part + (size_t)(blk * NHEAD + g) * 6;
    s += p[2 * t]; q += p[2 * t + 1];
  }
  const float cnt  = (float)BATCH * (float)(KW * KW);
  const float mean = s / cnt;
  const float var  = q / cnt - mean * mean;
  const float sc   = gma[ch] * rsqrtf(var + EPSB);
  bns[2 * ch]     = sc;
  bns[2 * ch + 1] = bta[ch] - mean * sc;
}

// ---------------------------------------------------------------------------
// K5: full attention per (b,g): qk(WMMA) -> +qr/kr (BN'd) -> softmax ->
// sv (WMMA, K=56) & sve -> so (channels interleaved sv/sve).
// ---------------------------------------------------------------------------
__global__ __launch_bounds__(128)
void k5_attention(const float* __restrict__ qkv, const float* __restrict__ bnq,
                  const float* __restrict__ bns, const float* __restrict__ relative,
                  float* __restrict__ so) {
  __shared__ float qs[4 * 64];
  __shared__ float ks[4 * 64];
  __shared__ float vs[16 * 64];     // rows 8..15 zero -> D rows 8..15 zero
  __shared__ float rl[16 * RELW];
  __shared__ float sim[64 * 68];    // stride 68 to dodge bank conflicts
  const int tid = threadIdx.x;
  const int bg = blockIdx.x;
  const int b = bg >> 3, g = bg & 7;

  for (int idx = tid; idx < 4 * 64; idx += 128) {
    const int c = idx >> 6, i = idx & 63;
    const int oq = g * 16 + c, ok = g * 16 + 4 + c;
    float vq = 0.f, vk = 0.f;
    if (i < KW) {
      vq = bnq[2 * oq] * qkv[(size_t)oq * PTOT + b * KW + i] + bnq[2 * oq + 1];
      vk = bnq[2 * ok] * qkv[(size_t)ok * PTOT + b * KW + i] + bnq[2 * ok + 1];
    }
    qs[idx] = vq; ks[idx] = vk;
  }
  for (int idx = tid; idx < 16 * 64; idx += 128) {
    const int c = idx >> 6, j = idx & 63;
    float vv = 0.f;
    if (c < 8 && j < KW) {
      const int ov = g * 16 + 8 + c;
      vv = bnq[2 * ov] * qkv[(size_t)ov * PTOT + b * KW + j] + bnq[2 * ov + 1];
    }
    vs[idx] = vv;
  }
  for (int idx = tid; idx < 16 * RELW; idx += 128) rl[idx] = relative[idx];
  __syncthreads();

  const float sc0 = bns[2 * g],            sh0 = bns[2 * g + 1];
  const float sc1 = bns[2 * (8 + g)],      sh1 = bns[2 * (8 + g) + 1];
  const float sc2 = bns[2 * (16 + g)],     sh2 = bns[2 * (16 + g) + 1];

  const int lane = tid & 31, wave = tid >> 5;
  const int mn = lane & 15, kh = lane >> 4;

  { // qk tiles -> sim (BN applied)
    const int j0 = wave * 16;
    for (int ti = 0; ti < 4; ++ti) {
      const int i0 = ti * 16;
      v2f a, bb;
      a.x  = qs[(2 * kh + 0) * 64 + i0 + mn];
      a.y  = qs[(2 * kh + 1) * 64 + i0 + mn];
      bb.x = ks[(2 * kh + 0) * 64 + j0 + mn];
      bb.y = ks[(2 * kh + 1) * 64 + j0 + mn];
      v8f d = {};
      d = wmma_f32_k4(a, bb, d);
      #pragma unroll
      for (int v = 0; v < 8; ++v) {
        const int m = v + 8 * kh;
        sim[(i0 + m) * 68 + j0 + mn] = sc0 * d[v] + sh0;
      }
    }
  }
  __syncthreads();

  for (int e = tid; e < KW * KW; e += 128) {
    const int i = e / KW, j = e - i * KW;
    const int dij = i - j + (KW - 1);
    const int dji = 110 - dij;
    float qr = qs[0 * 64 + i] * rl[0 * RELW + dij]
             + qs[1 * 64 + i] * rl[1 * RELW + dij]
             + qs[2 * 64 + i] * rl[2 * RELW + dij]
             + qs[3 * 64 + i] * rl[3 * RELW + dij];
    float kr = ks[0 * 64 + j] * rl[4 * RELW + dji]
             + ks[1 * 64 + j] * rl[5 * RELW + dji]
             + ks[2 * 64 + j] * rl[6 * RELW + dji]
             + ks[3 * 64 + j] * rl[7 * RELW + dji];
    sim[i * 68 + j] += (sc1 * (F_QR * qr) + sh1) + (sc2 * (F_KR * kr) + sh2);
  }
  __syncthreads();

  if (tid < KW) {  // softmax over j for row i = tid
    const int i = tid;
    float mx = -3.402823466e38f;
    for (int j = 0; j < KW; ++j) mx = fmaxf(mx, sim[i * 68 + j]);
    float s = 0.f;
    for (int j = 0; j < KW; ++j) {
      const float e2 = __expf(sim[i * 68 + j] - mx);
      sim[i * 68 + j] = e2; s += e2;
    }
    const float inv = 1.f / s;
    for (int j = 0; j < KW; ++j) sim[i * 68 + j] *= inv;
  }
  __syncthreads();

  { // sv[c][i] = sum_j v[c][j] * sim[i][j]  (M=c, N=i per wave, K=j=56)
    const int i0 = wave * 16;
    v8f acc = {};
    for (int js = 0; js < 14; ++js) {
      const int kk = js * 4 + 2 * kh;
      v2f a, bb;
      a.x  = vs[mn * 64 + kk];              // A[m=c][k=j]
      a.y  = vs[mn * 64 + kk + 1];
      bb.x = sim[(i0 + mn) * 68 + kk];      // B[k=j][n=i]
      bb.y = sim[(i0 + mn) * 68 + kk + 1];
      acc = wmma_f32_k4(a, bb, acc);
    }
    if (kh == 0) {                          // D rows 0..7 hold real channels
      #pragma unroll
      for (int v = 0; v < 8; ++v) {
        const int i = i0 + mn;
        if (i < KW)
          so[(size_t)(g * 16 + 2 * v) * PTOT + b * KW + i] = F_SV * acc[v];
      }
    }
  }

  for (int e = tid; e < 8 * KW; e += 128) {  // sve
    const int c = e / KW, i = e - c * KW;
    float s = 0.f;
    for (int j = 0; j < KW; ++j)
      s += sim[i * 68 + j] * rl[(8 + c) * RELW + (i - j + KW - 1)];
    so[(size_t)(g * 16 + 2 * c + 1) * PTOT + b * KW + i] = F_SVE * s;
  }
}

// ---------------------------------------------------------------------------
// K7: out[c][p] = BN(so[2c][p]) + BN(so[2c+1][p])  (output is (1,64,32,56,56))
// ---------------------------------------------------------------------------
__global__ __launch_bounds__(256)
void k7_output(const float* __restrict__ so, const float* __restrict__ bno,
               float* __restrict__ out) {
  const size_t idx = (size_t)blockIdx.x * 256 + threadIdx.x;
  const int c = (int)(idx / PTOT);
  const int p = (int)(idx - (size_t)c * PTOT);
  const int o0 = 2 * c, o1 = 2 * c + 1;
  const float a  = bno[2 * o0] * so[(size_t)o0 * PTOT + p] + bno[2 * o0 + 1];
  const float b2 = bno[2 * o1] * so[(size_t)o1 * PTOT + p] + bno[2 * o1 + 1];
  out[idx] = a + b2;
}

// ---------------------------------------------------------------------------
// Workspace layout (floats): qkv(51MB) | so(51MB) | bnq | partials | bns | bno
// ---------------------------------------------------------------------------
static constexpr size_t OFF_QKV  = 0;
static constexpr size_t OFF_SO   = OFF_QKV + (size_t)OC2 * PTOT;
static constexpr size_t OFF_BNQ  = OFF_SO + (size_t)OC2 * PTOT;
static constexpr size_t OFF_PART = OFF_BNQ + 2 * OC2;
static constexpr size_t OFF_BNS  = OFF_PART + (size_t)BATCH * NHEAD * 6;
static constexpr size_t OFF_BNO  = OFF_BNS + 2 * 24;

extern "C" void kernel_launch(void* const* d_in, const int* in_sizes, int n_in,
                              void* d_out, int out_size, void* d_ws, size_t ws_size,
                              hipStream_t stream) {
  (void)in_sizes; (void)n_in; (void)out_size; (void)ws_size;
  const float* x        = (const float*)d_in[0];
  const float* w_qkv    = (const float*)d_in[1];
  const float* relative = (const float*)d_in[2];
  const float* bn_qkv_g = (const float*)d_in[3];
  const float* bn_qkv_b = (const float*)d_in[4];
  const float* bn_sim_g = (const float*)d_in[5];
  const float* bn_sim_b = (const float*)d_in[6];
  const float* bn_out_g = (const float*)d_in[7];
  const float* bn_out_b = (const float*)d_in[8];
  float* out = (float*)d_out;
  float* ws  = (float*)d_ws;

  float* qkv  = ws + OFF_QKV;
  float* so   = ws + OFF_SO;
  float* bnq  = ws + OFF_BNQ;
  float* part = ws + OFF_PART;
  float* bns  = ws + OFF_BNS;
  float* bno  = ws + OFF_BNO;

  k1_qkv_gemm<<<dim3(PTOT / 64, OC2 / 16), 128, 0, stream>>>(x, w_qkv, qkv);
  k2_chstats<<<OC2, 256, 0, stream>>>(qkv, bn_qkv_g, bn_qkv_b, bnq);
  k3_sim_stats<<<BATCH * NHEAD, 128, 0, stream>>>(qkv, bnq, relative, part);
  k4_bns<<<1, 32, 0, stream>>>(part, bn_sim_g, bn_sim_b, bns);
  k5_attention<<<BATCH * NHEAD, 128, 0, stream>>>(qkv, bnq, bns, relative, so);
  k2_chstats<<<OC2, 256, 0, stream>>>(so, bn_out_g, bn_out_b, bno);
  k7_output<<<(64 * PTOT) / 256, 256, 0, stream>>>(so, bno, out);
}